// LengthRegulator_81071802679498
// MI455X (gfx1250) — compile-verified
//
#include <hip/hip_runtime.h>
#include <hip/hip_bf16.h>
#include <stdint.h>

// Problem shape (from reference setup_inputs): x:(32,512,256) f32, duration:(32,512) i32, max_len=4096.
#define BATCH    32
#define SLEN     512
#define DMODEL   256
#define MAXLEN   4096
#define ROWS_PER_BLOCK 8   // 8 rows * 64 lanes (one 16B vector each) = 512 threads

typedef __attribute__((ext_vector_type(4))) float v4f;   // native vector: OK for nontemporal builtins

// ---------------------------------------------------------------------------
// Kernel 1: per-batch inclusive scan of durations.
// wave32 shuffle scan (5 steps) + 16-wave-sum scan in wave 0; 2 barriers total.
// Writes cum (int32, SLEN per batch) to workspace and mel_len to output tail.
// ---------------------------------------------------------------------------
__global__ __launch_bounds__(SLEN) void lr_scan_kernel(const int* __restrict__ dur,
                                                       int* __restrict__ cum_ws,
                                                       float* __restrict__ mel_out) {
    __shared__ int wsum[SLEN / 32];            // 16 per-wave sums
    const int b    = blockIdx.x;
    const int tid  = threadIdx.x;
    const int lane = tid & 31;
    const int wave = tid >> 5;                 // 0..15

    int v = dur[b * SLEN + tid];
    v = v > 0 ? v : 0;                         // jnp.maximum(duration, 0)

    // Inclusive scan within the wave (wave32).
    int s = v;
    #pragma unroll
    for (int off = 1; off < 32; off <<= 1) {
        int n = __shfl_up(s, off, 32);
        if (lane >= off) s += n;
    }
    if (lane == 31) wsum[wave] = s;
    __syncthreads();

    // Wave 0 scans the 16 wave sums (inclusive).
    if (wave == 0 && lane < SLEN / 32) {
        int ws = wsum[lane];
        #pragma unroll
        for (int off = 1; off < SLEN / 32; off <<= 1) {
            int n = __shfl_up(ws, off, 32);
            if (lane >= off) ws += n;
        }
        wsum[lane] = ws;
    }
    __syncthreads();

    const int base = (wave > 0) ? wsum[wave - 1] : 0;
    const int cum  = base + s;

    cum_ws[b * SLEN + tid] = cum;
    if (tid == SLEN - 1) mel_out[b] = (float)cum;   // mel_len (exact in f32, <= 3584)
}

// ---------------------------------------------------------------------------
// Kernel 2: expansion. One block = 8 output rows of one batch.
//  - cum[b][0..511] pulled into LDS via async global->LDS b128 (ASYNCcnt path).
//  - branchless searchsorted in LDS (broadcast reads).
//  - 64 lanes per row: coalesced b128 loads; NT b128 stores so the 128MB
//    write-once stream doesn't evict the L2-resident x (16MB) / cum tiles.
// ---------------------------------------------------------------------------
__global__ __launch_bounds__(512) void lr_expand_kernel(const float* __restrict__ x,
                                                        const int* __restrict__ cum_ws,
                                                        float* __restrict__ out) {
    __shared__ int cums[SLEN];
    const int tid = threadIdx.x;
    const int blocks_per_batch = MAXLEN / ROWS_PER_BLOCK;   // 512
    const int b     = blockIdx.x / blocks_per_batch;
    const int tbase = (blockIdx.x % blocks_per_batch) * ROWS_PER_BLOCK;

    // Async copy of the 2KB cum tile into LDS: 128 lanes x 16B.
    // Generic->LDS address is the low 32 bits of the generic pointer on AMDGPU.
    if (tid < SLEN / 4) {
        unsigned lds_off = (unsigned)(uintptr_t)cums + (unsigned)tid * 16u;
        const int* gsrc = cum_ws + (size_t)b * SLEN + (size_t)tid * 4;
        asm volatile("global_load_async_to_lds_b128 %0, %1, off"
                     :
                     : "v"(lds_off), "v"(gsrc)
                     : "memory");
    }
    asm volatile("s_wait_asynccnt 0" ::: "memory");
    __syncthreads();

    const int row  = tid >> 6;       // 0..7
    const int lane = tid & 63;       // 16B-vector index within the 256-float row
    const int t    = tbase + row;
    const int mel  = cums[SLEN - 1];

    // Branchless upper_bound over cum (monotone, 512 entries -> 9 steps).
    // Result == min(searchsorted(cum, t, 'right'), SLEN-1) for all kept t.
    int idx = 0;
    #pragma unroll
    for (int w = SLEN >> 1; w > 0; w >>= 1) {
        if (cums[idx + w - 1] <= t) idx += w;
    }

    v4f v = (v4f)(0.0f);
    if (t < mel) {
        const v4f* src = (const v4f*)(x + ((size_t)b * SLEN + (size_t)idx) * DMODEL);
        v = src[lane];
    }
    v4f* dst = (v4f*)(out + ((size_t)b * MAXLEN + (size_t)t) * DMODEL);
    __builtin_nontemporal_store(v, dst + lane);
}

// ---------------------------------------------------------------------------
extern "C" void kernel_launch(void* const* d_in, const int* in_sizes, int n_in,
                              void* d_out, int out_size, void* d_ws, size_t ws_size,
                              hipStream_t stream) {
    const float* x  = (const float*)d_in[0];
    const int* dur  = (const int*)d_in[1];
    // d_in[2] = max_len (compile-time constant 4096 for this shape)

    float* out     = (float*)d_out;
    float* mel_out = out + (size_t)BATCH * MAXLEN * DMODEL;   // second tuple output
    int*   cum_ws  = (int*)d_ws;                              // 32*512*4 = 64 KB scratch

    lr_scan_kernel<<<BATCH, SLEN, 0, stream>>>(dur, cum_ws, mel_out);

    lr_expand_kernel<<<BATCH * (MAXLEN / ROWS_PER_BLOCK), 512, 0, stream>>>(x, cum_ws, out);
}